// PsiNN1_84688165142619
// MI455X (gfx1250) — compile-verified
//
#include <hip/hip_runtime.h>
#include <hip/hip_bf16.h>
#include <math.h>

// Problem dims
#define S_STEPS 16384
#define I_DIM   64
#define R_DIM   2048
#define H_DIM   128
#define O_DIM   50
#define ALPHA_C 0.5f

// Persistent scan config
#define NB          128   // workgroups in the scan grid
#define ROWS_PER_WG (R_DIM / NB)   // 16 rows of W per workgroup
#define TB          256   // threads per block (8 wave32)

typedef float v2f  __attribute__((ext_vector_type(2)));
typedef float v8f  __attribute__((ext_vector_type(8)));
typedef unsigned int u32x4 __attribute__((ext_vector_type(4)));
typedef int   i32x4 __attribute__((ext_vector_type(4)));
typedef int   i32x8 __attribute__((ext_vector_type(8)));

// ---------------------------------------------------------------------------
// Init: zero h double-buffer and grid-sync counter (graph-replay safe).
// ---------------------------------------------------------------------------
__global__ void esn_init_kernel(float* __restrict__ hbuf, unsigned int* __restrict__ syncc) {
    int tid = blockIdx.x * blockDim.x + threadIdx.x;
    if (tid < 2 * R_DIM) hbuf[tid] = 0.0f;
    if (tid == 0) *syncc = 0u;
}

// ---------------------------------------------------------------------------
// TDM helper: issue a 1-row (1 x elems) fp32 tile load global -> LDS.
// D# built per CDNA5 ISA section 8.3/8.4. Returns via TENSORcnt.
// ---------------------------------------------------------------------------
__device__ __forceinline__ void tdm_load_row_to_lds(const float* gsrc,
                                                    unsigned int lds_byte_off,
                                                    unsigned int elems) {
#if defined(__has_builtin) && __has_builtin(__builtin_amdgcn_tensor_load_to_lds)
    unsigned long long ga = (unsigned long long)(uintptr_t)gsrc;
    u32x4 g0;
    g0[0] = 0x1u;                                   // count=1, user mode
    g0[1] = lds_byte_off;                           // lds_addr
    g0[2] = (unsigned int)(ga & 0xFFFFFFFFu);       // global_addr[31:0]
    g0[3] = (unsigned int)((ga >> 32) & 0x1FFFFFFu) // global_addr[56:32]
          | (2u << 30);                             // type = 2 (image)
    i32x8 g1;
    g1[0] = (int)(2u << 16);                        // data_size=2 (4 bytes), mask=0
    g1[1] = (int)((elems & 0xFFFFu) << 16);         // tensor_dim0[15:0] at bits 63:48
    g1[2] = (int)(((elems >> 16) & 0xFFFFu)         // tensor_dim0[31:16] at bits 79:64
          | (1u << 16));                            // tensor_dim1 = 1 (bits 95:80)
    g1[3] = (int)((elems & 0xFFFFu) << 16);         // tile_dim0 at bits 127:112
    g1[4] = 1;                                      // tile_dim1 = 1
    g1[5] = (int)elems;                             // tensor_dim0_stride[31:0]
    g1[6] = 0;
    g1[7] = 0;
    i32x4 gz = {0, 0, 0, 0};
#if __clang_major__ >= 23
    i32x8 gz8 = {0, 0, 0, 0, 0, 0, 0, 0};
    __builtin_amdgcn_tensor_load_to_lds(g0, g1, gz, gz, gz8, 0);
#else
    __builtin_amdgcn_tensor_load_to_lds(g0, g1, gz, gz, 0);
#endif
#else
    (void)gsrc; (void)lds_byte_off; (void)elems;
#endif
}

// ---------------------------------------------------------------------------
// Phase 1: xin = x @ Win^T   (S x I)@(I x R) -> (S x R), fp32 WMMA 16x16x4.
// Each wave computes a 16(M) x 64(N) C strip: A fragment reused across 4 tiles.
// ---------------------------------------------------------------------------
__global__ void __launch_bounds__(TB)
xin_gemm_kernel(const float* __restrict__ x,    // S x I
                const float* __restrict__ Win,  // R x I
                float* __restrict__ xin)        // S x R
{
    const int lane = threadIdx.x & 31;
    const int wave = threadIdx.x >> 5;
    const int wgid = blockIdx.x * 8 + wave;
    const int nsup = R_DIM / 64;                      // 32 N-strips
    const int m0   = (wgid / nsup) * 16;
    const int n0   = (wgid % nsup) * 64;

    const int mn = lane & 15;
    const int kk = (lane >> 4) * 2;

    v8f c0 = {}, c1 = {}, c2 = {}, c3 = {};
    #pragma unroll
    for (int k0 = 0; k0 < I_DIM; k0 += 4) {
        v2f a, b0, b1, b2, b3;
        const float* ap = x + (size_t)(m0 + mn) * I_DIM + k0 + kk;
        a.x = ap[0]; a.y = ap[1];
        const float* bp = Win + (size_t)(n0 + mn) * I_DIM + k0 + kk;
        b0.x = bp[0];            b0.y = bp[1];
        b1.x = bp[16 * I_DIM];   b1.y = bp[16 * I_DIM + 1];
        b2.x = bp[32 * I_DIM];   b2.y = bp[32 * I_DIM + 1];
        b3.x = bp[48 * I_DIM];   b3.y = bp[48 * I_DIM + 1];
        c0 = __builtin_amdgcn_wmma_f32_16x16x4_f32(false, a, false, b0, (short)0, c0, false, false);
        c1 = __builtin_amdgcn_wmma_f32_16x16x4_f32(false, a, false, b1, (short)0, c1, false, false);
        c2 = __builtin_amdgcn_wmma_f32_16x16x4_f32(false, a, false, b2, (short)0, c2, false, false);
        c3 = __builtin_amdgcn_wmma_f32_16x16x4_f32(false, a, false, b3, (short)0, c3, false, false);
    }
    // C/D layout: VGPR v, lane L -> M = v + 8*(L>=16), N = L&15
    const int nl    = lane & 15;
    const int mbase = m0 + 8 * (lane >> 4);
    #pragma unroll
    for (int v = 0; v < 8; ++v) {
        float* orow = xin + (size_t)(mbase + v) * R_DIM + n0 + nl;
        orow[0]  = c0[v];
        orow[16] = c1[v];
        orow[32] = c2[v];
        orow[48] = c3[v];
    }
}

// ---------------------------------------------------------------------------
// Phase 2: persistent leaky-ESN scan.
// h_{t+1} = (1-a)h_t + a*tanh(xin[t] + W h_t); states[t] = h_{t+1}.
// 128 WGs x 16 rows each; h staged into LDS via TDM (tensor_load_to_lds);
// double-buffered h in global; device-wide release/acquire barrier per step.
// ---------------------------------------------------------------------------
__global__ void __launch_bounds__(TB)
esn_scan_kernel(const float* __restrict__ xin,     // S x R
                const float* __restrict__ W,       // R x R
                float* __restrict__ states,        // S x R
                float* __restrict__ hbuf,          // 2 x R
                unsigned int* __restrict__ syncc)
{
    __shared__ float h_lds[R_DIM];
    __shared__ float y_lds[ROWS_PER_WG];

    const int tid  = threadIdx.x;
    const int wave = tid >> 5;
    const int lane = tid & 31;
    const int row_base = blockIdx.x * ROWS_PER_WG;

    // LDS byte offset of h_lds: flat LDS aperture keeps the offset in addr[31:0]
    const unsigned int h_lds_off = (unsigned int)(uintptr_t)(void*)h_lds;

    // each wave owns 2 of the 16 rows
    const float* Wr0 = W + (size_t)(row_base + 2 * wave) * R_DIM;
    const float* Wr1 = Wr0 + R_DIM;

    for (int t = 0; t < S_STEPS; ++t) {
        const float* hsrc = hbuf + ((t) & 1) * R_DIM;
        float*       hdst = hbuf + ((t + 1) & 1) * R_DIM;

        // prefetch upcoming xin rows (global_prefetch_b8)
        if (tid < ROWS_PER_WG) {
            int tp = t + 4 < S_STEPS ? t + 4 : S_STEPS - 1;
            __builtin_prefetch(&xin[(size_t)tp * R_DIM + row_base + tid], 0, 0);
        }

#if defined(__has_builtin) && __has_builtin(__builtin_amdgcn_tensor_load_to_lds)
        // Stage h_t into LDS with the Tensor Data Mover (wave 0 issues the DMA)
        if (wave == 0) {
            tdm_load_row_to_lds(hsrc, h_lds_off, R_DIM);
            __builtin_amdgcn_s_wait_tensorcnt(0);
        }
#else
        for (int j = tid; j < R_DIM; j += TB) h_lds[j] = hsrc[j];
#endif
        __syncthreads();

        // partial matvec: lane accumulates strided columns for 2 rows
        float acc0 = 0.0f, acc1 = 0.0f;
        #pragma unroll 8
        for (int i = 0; i < R_DIM / 32; ++i) {
            int j = lane + (i << 5);
            float hj = h_lds[j];
            acc0 = fmaf(Wr0[j], hj, acc0);
            acc1 = fmaf(Wr1[j], hj, acc1);
        }
        // wave32 reduction
        #pragma unroll
        for (int off = 16; off > 0; off >>= 1) {
            acc0 += __shfl_down(acc0, off, 32);
            acc1 += __shfl_down(acc1, off, 32);
        }
        if (lane == 0) { y_lds[2 * wave] = acc0; y_lds[2 * wave + 1] = acc1; }
        __syncthreads();

        // leaky update + write-out for owned rows
        if (tid < ROWS_PER_WG) {
            int r = row_base + tid;
            float pre  = xin[(size_t)t * R_DIM + r] + y_lds[tid];
            float hnew = (1.0f - ALPHA_C) * h_lds[r] + ALPHA_C * tanhf(pre);
            hdst[r] = hnew;
            states[(size_t)t * R_DIM + r] = hnew;
            __threadfence();   // make writes device-visible before signaling
        }
        __syncthreads();

        // grid-wide split barrier: release add, acquire spin
        if (tid == 0) {
            __hip_atomic_fetch_add(syncc, 1u, __ATOMIC_ACQ_REL, __HIP_MEMORY_SCOPE_AGENT);
            unsigned int target = (unsigned int)NB * (unsigned int)(t + 1);
            while (__hip_atomic_load(syncc, __ATOMIC_ACQUIRE, __HIP_MEMORY_SCOPE_AGENT) < target)
                __builtin_amdgcn_s_sleep(2);
        }
        __syncthreads();
    }
}

// ---------------------------------------------------------------------------
// Phase 3a: koopman = states @ fc1_w^T + fc1_b  (S x R)@(R x H) -> (S x H).
// fp32 WMMA 16x16x4, K = 2048. Each wave: 16(M) x 64(N) strip, A reused 4x.
// ---------------------------------------------------------------------------
__global__ void __launch_bounds__(TB)
fc1_gemm_kernel(const float* __restrict__ states,  // S x R
                const float* __restrict__ fc1_w,   // H x R
                const float* __restrict__ fc1_b,   // H
                float* __restrict__ koop)          // S x H
{
    const int lane = threadIdx.x & 31;
    const int wave = threadIdx.x >> 5;
    const int wgid = blockIdx.x * 8 + wave;
    const int nsup = H_DIM / 64;                    // 2 N-strips
    const int m0   = (wgid / nsup) * 16;
    const int n0   = (wgid % nsup) * 64;

    const int mn = lane & 15;
    const int kk = (lane >> 4) * 2;

    v8f c0 = {}, c1 = {}, c2 = {}, c3 = {};
    for (int k0 = 0; k0 < R_DIM; k0 += 4) {
        v2f a, b0, b1, b2, b3;
        const float* ap = states + (size_t)(m0 + mn) * R_DIM + k0 + kk;
        a.x = ap[0]; a.y = ap[1];
        const float* bp = fc1_w + (size_t)(n0 + mn) * R_DIM + k0 + kk;
        b0.x = bp[0];            b0.y = bp[1];
        b1.x = bp[16 * R_DIM];   b1.y = bp[16 * R_DIM + 1];
        b2.x = bp[32 * R_DIM];   b2.y = bp[32 * R_DIM + 1];
        b3.x = bp[48 * R_DIM];   b3.y = bp[48 * R_DIM + 1];
        c0 = __builtin_amdgcn_wmma_f32_16x16x4_f32(false, a, false, b0, (short)0, c0, false, false);
        c1 = __builtin_amdgcn_wmma_f32_16x16x4_f32(false, a, false, b1, (short)0, c1, false, false);
        c2 = __builtin_amdgcn_wmma_f32_16x16x4_f32(false, a, false, b2, (short)0, c2, false, false);
        c3 = __builtin_amdgcn_wmma_f32_16x16x4_f32(false, a, false, b3, (short)0, c3, false, false);
    }
    const int nl    = lane & 15;
    const int mbase = m0 + 8 * (lane >> 4);
    const float bias0 = fc1_b[n0 + nl];
    const float bias1 = fc1_b[n0 + 16 + nl];
    const float bias2 = fc1_b[n0 + 32 + nl];
    const float bias3 = fc1_b[n0 + 48 + nl];
    #pragma unroll
    for (int v = 0; v < 8; ++v) {
        float* orow = koop + (size_t)(mbase + v) * H_DIM + n0 + nl;
        orow[0]  = c0[v] + bias0;
        orow[16] = c1[v] + bias1;
        orow[32] = c2[v] + bias2;
        orow[48] = c3[v] + bias3;
    }
}

// ---------------------------------------------------------------------------
// Phase 3b: out = koopman @ fc2_w^T + fc2_b  (S x 128)@(128 x 50).
// ---------------------------------------------------------------------------
__global__ void __launch_bounds__(H_DIM)
fc2_kernel(const float* __restrict__ koop,   // S x H
           const float* __restrict__ fc2_w,  // O x H
           const float* __restrict__ fc2_b,  // O
           float* __restrict__ out)          // S x O
{
    __shared__ float krow[H_DIM];
    const int s   = blockIdx.x;
    const int tid = threadIdx.x;
    krow[tid] = koop[(size_t)s * H_DIM + tid];
    __syncthreads();
    if (tid < O_DIM) {
        const float* w = fc2_w + (size_t)tid * H_DIM;
        float acc = fc2_b[tid];
        #pragma unroll 8
        for (int k = 0; k < H_DIM; ++k) acc = fmaf(w[k], krow[k], acc);
        out[(size_t)s * O_DIM + tid] = acc;
    }
}

// ---------------------------------------------------------------------------
extern "C" void kernel_launch(void* const* d_in, const int* in_sizes, int n_in,
                              void* d_out, int out_size, void* d_ws, size_t ws_size,
                              hipStream_t stream) {
    const float* x     = (const float*)d_in[0];   // S x I
    const float* Win   = (const float*)d_in[1];   // R x I
    const float* W     = (const float*)d_in[2];   // R x R
    const float* fc1_w = (const float*)d_in[3];   // H x R
    const float* fc1_b = (const float*)d_in[4];   // H
    const float* fc2_w = (const float*)d_in[5];   // O x H
    const float* fc2_b = (const float*)d_in[6];   // O
    float* out = (float*)d_out;                   // S x O

    // Workspace layout
    float* xin    = (float*)d_ws;                              // S*R
    float* states = xin    + (size_t)S_STEPS * R_DIM;          // S*R
    float* koop   = states + (size_t)S_STEPS * R_DIM;          // S*H
    float* hbuf   = koop   + (size_t)S_STEPS * H_DIM;          // 2*R
    unsigned int* syncc = (unsigned int*)(hbuf + 2 * R_DIM);

    // init h double-buffer + sync counter
    esn_init_kernel<<<(2 * R_DIM + 255) / 256, 256, 0, stream>>>(hbuf, syncc);

    // Phase 1: input projection GEMM (WMMA, 16x64 strip per wave)
    {
        int waves = (S_STEPS / 16) * (R_DIM / 64);  // 32768
        xin_gemm_kernel<<<waves / 8, TB, 0, stream>>>(x, Win, xin);
    }

    // Phase 2: persistent sequential scan (TDM h-staging + grid barrier)
    esn_scan_kernel<<<NB, TB, 0, stream>>>(xin, W, states, hbuf, syncc);

    // Phase 3a: readout GEMM (WMMA, 16x64 strip per wave)
    {
        int waves = (S_STEPS / 16) * (H_DIM / 64);  // 2048
        fc1_gemm_kernel<<<waves / 8, TB, 0, stream>>>(states, fc1_w, fc1_b, koop);
    }

    // Phase 3b: final small GEMM
    fc2_kernel<<<S_STEPS, H_DIM, 0, stream>>>(koop, fc2_w, fc2_b, out);
}